// GroupNSTokenizer_4922032521978
// MI455X (gfx1250) — compile-verified
//
#include <hip/hip_runtime.h>
#include <math.h>

typedef __attribute__((ext_vector_type(2))) float v2f;
typedef __attribute__((ext_vector_type(8))) float v8f;

#define BROWS 16
#define CAT_STRIDE 516   // 512-col pad: (516 % 64) == 4 -> conflict-free A-column reads
#define DMODEL 512

struct Params {
  const int*   int_feats;       // [B,89] int32
  const float* emb[12];         // (vs+1, d) each
  const float* attn_w[3];       // fids 2,5,9 : (64,1)
  const float* attn_b[3];       // (1,)
  const float* gw1[4]; const float* gb1[4];   // (dim,h),(h,)
  const float* gw2[4]; const float* gb2[4];   // (h,dim),(dim,)
  const float* pw[4];  const float* pbias[4]; // (dim,512),(512,)
  const float* lng[4]; const float* lnb[4];   // (512,),(512,)
  float* out;                   // [B,4,512]
};

__global__ __launch_bounds__(256)
void tokenizer_fused_kernel(Params P) {
  constexpr int NF = 12;
  constexpr int F_OFF[NF]={0,1,2,22,23,24,74,75,76,77,87,88};
  constexpr int F_LEN[NF]={1,1,20,1,1,50,1,1,10,1,1,1};
  constexpr int F_DIM[NF]={64,32,64,16,64,64,8,4,32,64,32,64};
  constexpr int F_CAT[NF]={0,64,96,160,176,240,304,312,316,348,412,444};
  constexpr int MF[3]={2,5,9};
  constexpr int GD[4]={160,144,108,96};   // group dims (all % 4 == 0)
  constexpr int GH[4]={40,36,27,24};      // gate hidden
  constexpr int GB[4]={0,160,304,412};    // cat base per group
  constexpr int HB[4]={0,40,76,103};      // hidden base in s_hid row

  __shared__ float s_cat[BROWS][CAT_STRIDE];
  __shared__ float s_log[BROWS][52];
  __shared__ float s_hid[BROWS][128];
  __shared__ float s_sum[BROWS];
  __shared__ float s_sq[BROWS];

  const int t   = threadIdx.x;
  const int tr  = t >> 4;          // row within tile (16 rows x 16 subs)
  const int sub = t & 15;
  const int row0 = blockIdx.x * BROWS;
  const int* feats = P.int_feats + (long)(row0 + tr) * 89;

  // ---------------- Phase 1a: single-valued embedding gathers ----------------
  #pragma unroll
  for (int f = 0; f < NF; ++f) {
    if (F_LEN[f] == 1) {
      int idx = feats[F_OFF[f]];
      const float* tab = P.emb[f] + (long)idx * F_DIM[f];
      for (int j = sub; j < F_DIM[f]; j += 16)
        s_cat[tr][F_CAT[f] + j] = tab[j];
    }
  }

  // ---------------- Phase 1b: multi-valued attention pooling (d=64) ----------
  #pragma unroll
  for (int mi = 0; mi < 3; ++mi) {
    const int f = MF[mi];
    const int L = F_LEN[f];
    const float* w  = P.attn_w[mi];
    const float  bi = P.attn_b[mi][0];
    __syncthreads();                       // protect s_log reuse
    for (int l = sub; l < L; l += 16) {
      int v = feats[F_OFF[f] + l];
      const float* e = P.emb[f] + (long)v * 64;
      float acc = bi;
      #pragma unroll
      for (int j = 0; j < 64; ++j) acc += e[j] * w[j];
      s_log[tr][l] = (v != 0) ? acc : -1.0e9f;
    }
    __syncthreads();
    float mx = -INFINITY;
    for (int l = 0; l < L; ++l) mx = fmaxf(mx, s_log[tr][l]);
    float sm = 0.f;
    for (int l = 0; l < L; ++l) sm += expf(s_log[tr][l] - mx);
    float inv = 1.f / sm;
    float a0 = 0.f, a1 = 0.f, a2 = 0.f, a3 = 0.f;
    for (int l = 0; l < L; ++l) {
      float a = expf(s_log[tr][l] - mx) * inv;
      const float* e = P.emb[f] + (long)feats[F_OFF[f] + l] * 64;
      a0 += a * e[sub];       a1 += a * e[sub + 16];
      a2 += a * e[sub + 32];  a3 += a * e[sub + 48];
    }
    s_cat[tr][F_CAT[f] + sub]      = a0;
    s_cat[tr][F_CAT[f] + sub + 16] = a1;
    s_cat[tr][F_CAT[f] + sub + 32] = a2;
    s_cat[tr][F_CAT[f] + sub + 48] = a3;
  }
  __syncthreads();

  // ---------------- Phase 2: gate MLPs (per-row, VALU) -----------------------
  #pragma unroll
  for (int gi = 0; gi < 4; ++gi) {
    const int dim = GD[gi], h = GH[gi], base = GB[gi], hb = HB[gi];
    for (int j = sub; j < h; j += 16) {
      float acc = P.gb1[gi][j];
      for (int k = 0; k < dim; ++k) acc += s_cat[tr][base + k] * P.gw1[gi][k * h + j];
      s_hid[tr][hb + j] = acc / (1.f + expf(-acc));   // silu
    }
  }
  __syncthreads();
  #pragma unroll
  for (int gi = 0; gi < 4; ++gi) {
    const int dim = GD[gi], h = GH[gi], base = GB[gi], hb = HB[gi];
    for (int j = sub; j < dim; j += 16) {
      float acc = P.gb2[gi][j];
      for (int k = 0; k < h; ++k) acc += s_hid[tr][hb + k] * P.gw2[gi][k * dim + j];
      float g = 1.f / (1.f + expf(-acc));             // sigmoid
      s_cat[tr][base + j] *= (1.f + g);
    }
  }

  // ---------------- Phase 3: projection GEMM via V_WMMA_F32_16X16X4_F32 ------
  // Wave w owns output columns [w*64, w*64+64) as 4 C-tiles of 16x16.
  const int wave = t >> 5;
  const int lane = t & 31;
  const int half = lane >> 4;      // 0: K=k,k+1 ; 1: K=k+2,k+3 (ISA A/B layout)
  const int l16  = lane & 15;
  const int kh   = 2 * half;
  const int n0   = wave * 64 + l16;

  #pragma unroll
  for (int gi = 0; gi < 4; ++gi) {
    const int dim = GD[gi], base = GB[gi];
    const float* pw = P.pw[gi];
    const float* pb = P.pbias[gi];
    const float* g  = P.lng[gi];
    const float* be = P.lnb[gi];

    __syncthreads();                         // cat stable / prev-group stats read done
    if (t < BROWS) { s_sum[t] = 0.f; s_sq[t] = 0.f; }
    __syncthreads();

    v8f c0 = {}, c1 = {}, c2 = {}, c3 = {};
    for (int k = 0; k < dim; k += 4) {
      // A fragment: 16x4 of s_cat rows (lanes 0-15 rows M=0..15 hold K=k,k+1; lanes 16-31 K=k+2,k+3)
      v2f A; A.x = s_cat[l16][base + k + kh];
             A.y = s_cat[l16][base + k + kh + 1];
      const float* pr0 = pw + (long)(k + kh) * DMODEL;
      const float* pr1 = pw + (long)(k + kh + 1) * DMODEL;
      v2f B0; B0.x = pr0[n0];      B0.y = pr1[n0];
      v2f B1; B1.x = pr0[n0 + 16]; B1.y = pr1[n0 + 16];
      v2f B2; B2.x = pr0[n0 + 32]; B2.y = pr1[n0 + 32];
      v2f B3; B3.x = pr0[n0 + 48]; B3.y = pr1[n0 + 48];
      c0 = __builtin_amdgcn_wmma_f32_16x16x4_f32(false, A, false, B0, (short)0, c0, false, false);
      c1 = __builtin_amdgcn_wmma_f32_16x16x4_f32(false, A, false, B1, (short)0, c1, false, false);
      c2 = __builtin_amdgcn_wmma_f32_16x16x4_f32(false, A, false, B2, (short)0, c2, false, false);
      c3 = __builtin_amdgcn_wmma_f32_16x16x4_f32(false, A, false, B3, (short)0, c3, false, false);
    }

    // bias (per column)
    float pb0 = pb[n0], pb1 = pb[n0 + 16], pb2 = pb[n0 + 32], pb3 = pb[n0 + 48];
    #pragma unroll
    for (int v = 0; v < 8; ++v) { c0[v] += pb0; c1[v] += pb1; c2[v] += pb2; c3[v] += pb3; }

    // per-row sum / sumsq: reduce 16 columns in half-wave, accumulate across waves in LDS
    #pragma unroll
    for (int v = 0; v < 8; ++v) {
      float s = c0[v] + c1[v] + c2[v] + c3[v];
      float q = c0[v]*c0[v] + c1[v]*c1[v] + c2[v]*c2[v] + c3[v]*c3[v];
      #pragma unroll
      for (int m = 1; m < 16; m <<= 1) {
        s += __shfl_xor(s, m, 32);
        q += __shfl_xor(q, m, 32);
      }
      if (l16 == 0) {
        atomicAdd(&s_sum[v + 8 * half], s);   // ds_add_f32
        atomicAdd(&s_sq [v + 8 * half], q);
      }
    }
    __syncthreads();

    // layernorm + silu + store
    #pragma unroll
    for (int v = 0; v < 8; ++v) {
      int r = v + 8 * half;                              // C layout: VGPR v -> row v (lanes<16) / v+8
      float mean = s_sum[r] * (1.f / 512.f);
      float var  = s_sq[r] * (1.f / 512.f) - mean * mean;
      float rstd = rsqrtf(var + 1e-5f);
      float* orow = P.out + ((long)(row0 + r) * 4 + gi) * DMODEL;
      float y;
      y = (c0[v]-mean)*rstd*g[n0]      + be[n0];      orow[n0]      = y / (1.f + expf(-y));
      y = (c1[v]-mean)*rstd*g[n0+16]   + be[n0+16];   orow[n0+16]   = y / (1.f + expf(-y));
      y = (c2[v]-mean)*rstd*g[n0+32]   + be[n0+32];   orow[n0+32]   = y / (1.f + expf(-y));
      y = (c3[v]-mean)*rstd*g[n0+48]   + be[n0+48];   orow[n0+48]   = y / (1.f + expf(-y));
    }
  }
}

extern "C" void kernel_launch(void* const* d_in, const int* in_sizes, int n_in,
                              void* d_out, int out_size, void* d_ws, size_t ws_size,
                              hipStream_t stream) {
  // setup_inputs() flat order:
  // 0: int_feats | 1..12: embs[0..11] | 13..18: (attn_w,attn_b) for fids 2,5,9
  // 19..34: gate (w1,b1,w2,b2) x4 | 35..50: proj (pw,pb,g,be) x4
  Params P;
  int i = 0;
  P.int_feats = (const int*)d_in[i++];
  for (int f = 0; f < 12; ++f) P.emb[f] = (const float*)d_in[i++];
  for (int m = 0; m < 3; ++m) {
    P.attn_w[m] = (const float*)d_in[i++];
    P.attn_b[m] = (const float*)d_in[i++];
  }
  for (int g = 0; g < 4; ++g) {
    P.gw1[g] = (const float*)d_in[i++];
    P.gb1[g] = (const float*)d_in[i++];
    P.gw2[g] = (const float*)d_in[i++];
    P.gb2[g] = (const float*)d_in[i++];
  }
  for (int g = 0; g < 4; ++g) {
    P.pw[g]    = (const float*)d_in[i++];
    P.pbias[g] = (const float*)d_in[i++];
    P.lng[g]   = (const float*)d_in[i++];
    P.lnb[g]   = (const float*)d_in[i++];
  }
  P.out = (float*)d_out;

  const int B = in_sizes[0] / 89;          // 32768
  dim3 grid(B / BROWS), block(256);
  tokenizer_fused_kernel<<<grid, block, 0, stream>>>(P);
}